// _MultiheadAttention_9466107920857
// MI455X (gfx1250) — compile-verified
//
#include <hip/hip_runtime.h>
#include <hip/hip_bf16.h>
#include <math.h>

#define EMB 768
#define HEADS 12
#define HD 64
#define BS 4
#define SEQ 1024
#define BH (BS * HEADS)

typedef __attribute__((ext_vector_type(2))) float v2f;
typedef __attribute__((ext_vector_type(8))) float v8f;

__device__ __forceinline__ v8f wmma4(v2f a, v2f b, v8f c) {
    // D = A(16x4 f32) x B(4x16 f32) + C(16x16 f32)
    return __builtin_amdgcn_wmma_f32_16x16x4_f32(
        /*neg_a=*/false, a, /*neg_b=*/false, b,
        /*c_mod=*/(short)0, c, /*reuse_a=*/false, /*reuse_b=*/false);
}

// Store one 16x16 C tile. mbase = tile row origin, n = this lane's column.
template <bool HEAD_SPLIT>
__device__ __forceinline__ void store_proj_tile(float* __restrict__ Y, const v8f& c,
                                                int mbase, int n, float bn, int half) {
#pragma unroll
    for (int r = 0; r < 8; ++r) {
        const int m = mbase + half * 8 + r;
        const float val = c[r] + bn;
        if (HEAD_SPLIT) {
            const int b = m >> 10, s = m & (SEQ - 1);
            const int h = n >> 6, d = n & (HD - 1);
            Y[(((long)(b * HEADS + h)) * SEQ + s) * HD + d] = val;
        } else {
            Y[(long)m * EMB + n] = val;
        }
    }
}

// ---------------------------------------------------------------------------
// Y = X @ W^T + bias.   X:[4096,K] row-major, W:[768,K] row-major.
// Wave computes a 32x32 C block (2x2 WMMA tiles). Block = 4 waves -> 128 cols.
// grid = (4096/32, 768/128) = (128, 6), block = 128.
// ---------------------------------------------------------------------------
template <bool HEAD_SPLIT>
__global__ void proj_gemm(const float* __restrict__ X,
                          const float* __restrict__ W,
                          const float* __restrict__ bias,
                          float* __restrict__ Y, int K) {
    const int lane = threadIdx.x & 31;
    const int wave = threadIdx.x >> 5;
    const int half = lane >> 4;
    const int t    = lane & 15;
    const int m0 = blockIdx.x * 32;
    const int n0 = (blockIdx.y * 4 + wave) * 32;

    const float* a0 = X + (long)(m0 + t) * K + 2 * half;
    const float* a1 = a0 + (long)16 * K;
    const float* b0 = W + (long)(n0 + t) * K + 2 * half;
    const float* b1 = b0 + (long)16 * K;

    v8f c00 = {}, c01 = {}, c10 = {}, c11 = {};
#pragma unroll 8
    for (int kk = 0; kk < K; kk += 4) {
        const v2f ra0 = *(const v2f*)(a0 + kk);
        const v2f ra1 = *(const v2f*)(a1 + kk);
        const v2f rb0 = *(const v2f*)(b0 + kk);
        const v2f rb1 = *(const v2f*)(b1 + kk);
        c00 = wmma4(ra0, rb0, c00);
        c01 = wmma4(ra0, rb1, c01);
        c10 = wmma4(ra1, rb0, c10);
        c11 = wmma4(ra1, rb1, c11);
    }

    const float bn0 = bias[n0 + t];
    const float bn1 = bias[n0 + 16 + t];
    store_proj_tile<HEAD_SPLIT>(Y, c00, m0,      n0 + t,      bn0, half);
    store_proj_tile<HEAD_SPLIT>(Y, c01, m0,      n0 + 16 + t, bn1, half);
    store_proj_tile<HEAD_SPLIT>(Y, c10, m0 + 16, n0 + t,      bn0, half);
    store_proj_tile<HEAD_SPLIT>(Y, c11, m0 + 16, n0 + 16 + t, bn1, half);
}

__device__ __forceinline__ void store_score_tile(float* __restrict__ scores,
                                                 const float* __restrict__ prev,
                                                 long base, const v8f& c,
                                                 int mbase, int n, float scale, int half) {
#pragma unroll
    for (int r = 0; r < 8; ++r) {
        const int m = mbase + half * 8 + r;
        const long idx = base + (long)m * SEQ + n;
        scores[idx] = c[r] * scale + prev[idx];
    }
}

// ---------------------------------------------------------------------------
// scores[bh, m, n] = 0.125 * (q[bh,m,:] . k[bh,n,:]) + prev[bh, m, n]
// Wave = 32x32 block; block = 4 waves -> 128 key-cols.
// grid = (32, 8, 48), block = 128.
// ---------------------------------------------------------------------------
__global__ void attn_scores_kernel(const float* __restrict__ q,
                                   const float* __restrict__ k,
                                   const float* __restrict__ prev,
                                   float* __restrict__ scores, float scale) {
    const int lane = threadIdx.x & 31;
    const int wave = threadIdx.x >> 5;
    const int half = lane >> 4;
    const int t    = lane & 15;
    const int bh = blockIdx.z;
    const int m0 = blockIdx.x * 32;
    const int n0 = (blockIdx.y * 4 + wave) * 32;

    const float* qb = q + (long)bh * SEQ * HD;
    const float* kb = k + (long)bh * SEQ * HD;
    const float* a0 = qb + (long)(m0 + t) * HD + 2 * half;
    const float* a1 = a0 + 16 * HD;
    const float* b0 = kb + (long)(n0 + t) * HD + 2 * half;
    const float* b1 = b0 + 16 * HD;

    v8f c00 = {}, c01 = {}, c10 = {}, c11 = {};
#pragma unroll
    for (int kk = 0; kk < HD; kk += 4) {
        const v2f ra0 = *(const v2f*)(a0 + kk);
        const v2f ra1 = *(const v2f*)(a1 + kk);
        const v2f rb0 = *(const v2f*)(b0 + kk);
        const v2f rb1 = *(const v2f*)(b1 + kk);
        c00 = wmma4(ra0, rb0, c00);
        c01 = wmma4(ra0, rb1, c01);
        c10 = wmma4(ra1, rb0, c10);
        c11 = wmma4(ra1, rb1, c11);
    }

    const long base = (long)bh * SEQ * SEQ;
    store_score_tile(scores, prev, base, c00, m0,      n0 + t,      scale, half);
    store_score_tile(scores, prev, base, c01, m0,      n0 + 16 + t, scale, half);
    store_score_tile(scores, prev, base, c10, m0 + 16, n0 + t,      scale, half);
    store_score_tile(scores, prev, base, c11, m0 + 16, n0 + 16 + t, scale, half);
}

// ---------------------------------------------------------------------------
// Row softmax over 1024 cols. grid = 49152, block = 256 (4 elems/thread).
// ---------------------------------------------------------------------------
__global__ void softmax_kernel(const float* __restrict__ scores,
                               float* __restrict__ weights) {
    __shared__ float red[256];
    const long row = blockIdx.x;
    const float* in = scores + row * (long)SEQ;
    float* out = weights + row * (long)SEQ;
    const int tid = threadIdx.x;

    float v0 = in[tid];
    float v1 = in[tid + 256];
    float v2 = in[tid + 512];
    float v3 = in[tid + 768];

    float m = fmaxf(fmaxf(v0, v1), fmaxf(v2, v3));
    red[tid] = m;
    __syncthreads();
    for (int s = 128; s > 0; s >>= 1) {
        if (tid < s) red[tid] = fmaxf(red[tid], red[tid + s]);
        __syncthreads();
    }
    m = red[0];
    __syncthreads();

    v0 = expf(v0 - m);
    v1 = expf(v1 - m);
    v2 = expf(v2 - m);
    v3 = expf(v3 - m);

    red[tid] = v0 + v1 + v2 + v3;
    __syncthreads();
    for (int s = 128; s > 0; s >>= 1) {
        if (tid < s) red[tid] += red[tid + s];
        __syncthreads();
    }
    const float inv = 1.0f / red[0];

    out[tid]       = v0 * inv;
    out[tid + 256] = v1 * inv;
    out[tid + 512] = v2 * inv;
    out[tid + 768] = v3 * inv;
}

// ---------------------------------------------------------------------------
// tmp[b, s, h*64+d] = sum_k weights[bh, s, k] * v[bh, k, d]
// Wave = 32x32 block; block = 4 waves covering 64 rows x 64 cols.
// grid = (1024/64, 1, 48) = (16, 1, 48), block = 128.
// ---------------------------------------------------------------------------
__global__ void attn_av_kernel(const float* __restrict__ w,
                               const float* __restrict__ v,
                               float* __restrict__ tmp) {
    const int lane = threadIdx.x & 31;
    const int wave = threadIdx.x >> 5;
    const int half = lane >> 4;
    const int t    = lane & 15;
    const int bh = blockIdx.z;
    const int b  = bh / HEADS;
    const int h  = bh % HEADS;
    const int m0 = blockIdx.x * 64 + (wave >> 1) * 32;
    const int n0 = (wave & 1) * 32;

    const float* A = w + (long)bh * SEQ * SEQ;
    const float* B = v + (long)bh * SEQ * HD;
    const float* a0 = A + (long)(m0 + t) * SEQ + 2 * half;
    const float* a1 = a0 + (long)16 * SEQ;
    const float* bc0 = B + n0 + t + (long)(2 * half) * HD;
    const float* bc1 = bc0 + 16;

    v8f c00 = {}, c01 = {}, c10 = {}, c11 = {};
#pragma unroll 4
    for (int kk = 0; kk < SEQ; kk += 4) {
        const v2f ra0 = *(const v2f*)(a0 + kk);
        const v2f ra1 = *(const v2f*)(a1 + kk);
        v2f rb0, rb1;
        const long o = (long)kk * HD;
        rb0.x = bc0[o];
        rb0.y = bc0[o + HD];
        rb1.x = bc1[o];
        rb1.y = bc1[o + HD];
        c00 = wmma4(ra0, rb0, c00);
        c01 = wmma4(ra0, rb1, c01);
        c10 = wmma4(ra1, rb0, c10);
        c11 = wmma4(ra1, rb1, c11);
    }

    const int e0 = h * HD + n0 + t;
#pragma unroll
    for (int r = 0; r < 8; ++r) {
        const int s0 = m0 + half * 8 + r;
        tmp[((long)b * SEQ + s0) * EMB + e0]           = c00[r];
        tmp[((long)b * SEQ + s0) * EMB + e0 + 16]      = c01[r];
        tmp[((long)b * SEQ + s0 + 16) * EMB + e0]      = c10[r];
        tmp[((long)b * SEQ + s0 + 16) * EMB + e0 + 16] = c11[r];
    }
}

// ---------------------------------------------------------------------------
extern "C" void kernel_launch(void* const* d_in, const int* in_sizes, int n_in,
                              void* d_out, int out_size, void* d_ws, size_t ws_size,
                              hipStream_t stream) {
    (void)in_sizes; (void)n_in; (void)out_size; (void)ws_size;

    const float* Q    = (const float*)d_in[0];
    const float* Kin  = (const float*)d_in[1];
    const float* Vin  = (const float*)d_in[2];
    const float* prev = (const float*)d_in[3];
    const float* Wq   = (const float*)d_in[4];
    const float* bq   = (const float*)d_in[5];
    const float* Wk   = (const float*)d_in[6];
    const float* bk   = (const float*)d_in[7];
    const float* Wv   = (const float*)d_in[8];
    const float* bv   = (const float*)d_in[9];
    const float* Wo   = (const float*)d_in[10];
    const float* bo   = (const float*)d_in[11];

    const long NTOK = (long)BS * SEQ;          // 4096
    float* q_ws = (float*)d_ws;                // [BH, SEQ, HD]
    float* k_ws = q_ws + NTOK * EMB;
    float* v_ws = k_ws + NTOK * EMB;
    float* tmp  = q_ws;                        // reuse q region after scores

    float* out     = (float*)d_out;                          // [BS, SEQ, EMB]
    float* weights = out + NTOK * EMB;                       // [BH, SEQ, SEQ]
    float* scores  = weights + (long)BH * SEQ * SEQ;         // [BH, SEQ, SEQ]

    const dim3 blk(128);
    const dim3 gproj(NTOK / 32, EMB / 128);    // (128, 6)

    proj_gemm<true><<<gproj, blk, 0, stream>>>(Q,   Wq, bq, q_ws, EMB);
    proj_gemm<true><<<gproj, blk, 0, stream>>>(Kin, Wk, bk, k_ws, EMB);
    proj_gemm<true><<<gproj, blk, 0, stream>>>(Vin, Wv, bv, v_ws, EMB);

    attn_scores_kernel<<<dim3(SEQ / 32, SEQ / 128, BH), blk, 0, stream>>>(
        q_ws, k_ws, prev, scores, 0.125f);

    softmax_kernel<<<dim3(BH * SEQ), dim3(256), 0, stream>>>(scores, weights);

    attn_av_kernel<<<dim3(SEQ / 64, 1, BH), blk, 0, stream>>>(weights, v_ws, tmp);

    proj_gemm<false><<<gproj, blk, 0, stream>>>(tmp, Wo, bo, out, EMB);
}